// hyperRNN_7172595385052
// MI455X (gfx1250) — compile-verified
//
#include <hip/hip_runtime.h>
#include <cmath>

typedef float v2f __attribute__((ext_vector_type(2)));
typedef float v8f __attribute__((ext_vector_type(8)));

#define EPSF      1e-15f
#define PROJ_EPSF 1e-5f
#define HD   1024   // H*D
#define KCH  512    // K chunk held in LDS (32 KB)
#define NB   256    // batch
#define SQ   256    // sequence length

// sum over groups of 4 adjacent lanes (head dim D=4 maps to 4 adjacent N-lanes)
__device__ __forceinline__ float grp4_sum(float v) {
  v += __shfl_xor(v, 1, 32);
  v += __shfl_xor(v, 2, 32);
  return v;
}

__device__ __forceinline__ float atanh_ref(float x) {
  return 0.5f * logf((1.0f + x) / (1.0f - x));
}

// Computes one 16x16 tile of project( mob_mat_mul(M2, A-rows) ).
// A: 16 consecutive rows (stride HD). Tile columns [n0, n0+16).
// Output per lane: res[i] = row (i + 8*(lane>=16)), col n0+(lane&15)
// -- matches CDNA5 WMMA 32-bit C/D layout.
__device__ __forceinline__ void mmm_tile(
    const float* __restrict__ A, const float* __restrict__ M2,
    float* smem, int tid, int lane, int n0, float (&res)[8])
{
  const int mrow = (lane >> 4) * 8;   // C-tile row base for this lane
  const int ncol = lane & 15;
  const int nn   = n0 + ncol;
  v2f* sm2 = reinterpret_cast<v2f*>(smem);          // [KCH/2][16] paired-K, transposed
  const v2f* A2 = reinterpret_cast<const v2f*>(A);

  v8f c = {};
  float xpr[8] = {};

  for (int kc = 0; kc < HD; kc += KCH) {
    // cooperative load of A chunk, transposed + K-paired (coalesced global reads)
    for (int idx = tid; idx < 16 * (KCH / 2); idx += 256) {
      int r  = idx >> 8;        // KCH/2 == 256
      int k2 = idx & 255;
      sm2[k2 * 16 + r] = A2[(size_t)r * (HD / 2) + (kc >> 1) + k2];
    }
    __syncthreads();
    // in-place hyperbolic projection per head (4 consecutive K values)
    for (int hidx = tid; hidx < 16 * (KCH / 4); hidx += 256) {
      int r = hidx & 15;
      int j = hidx >> 4;
      v2f s01 = sm2[(2 * j) * 16 + r];
      v2f s23 = sm2[(2 * j + 1) * 16 + r];
      float ssq = s01.x * s01.x + s01.y * s01.y + s23.x * s23.x + s23.y * s23.y;
      float sc  = (1.0f - PROJ_EPSF) / fmaxf(ssq, 1.0f - PROJ_EPSF);
      sm2[(2 * j) * 16 + r]     = s01 * sc;
      sm2[(2 * j + 1) * 16 + r] = s23 * sc;
    }
    __syncthreads();
    // stash projected-input values needed later for x_norm (column nn of A)
    if (nn >= kc && nn < kc + KCH) {
      int p = (nn - kc) >> 1;
      #pragma unroll
      for (int i = 0; i < 8; ++i) {
        v2f q = sm2[p * 16 + (mrow + i)];
        xpr[i] = (nn & 1) ? q.y : q.x;
      }
    }
    // fp32 WMMA K-loop: D(16x16) += A(16x4) * B(4x16)
    #pragma unroll 8
    for (int k = 0; k < KCH; k += 4) {
      // A frag: VGPR0 = K {k, k+2} by half-wave; VGPR1 = +1  (paired in LDS)
      v2f a = sm2[((k >> 1) + (lane >> 4)) * 16 + ncol];
      // B frag: VGPR0 rows {k, k+1}; VGPR1 rows {k+2, k+3}; N = lane&15
      v2f bb;
      bb.x = M2[(size_t)(kc + k + (lane >> 4)) * HD + nn];
      bb.y = M2[(size_t)(kc + k + 2 + (lane >> 4)) * HD + nn];
      c = __builtin_amdgcn_wmma_f32_16x16x4_f32(
              false, a, false, bb, (short)0, c, false, false);
    }
    __syncthreads();
  }

  // mob_mat_mul epilogue: tanh(|Mx|/|x| * atanh(|x|)) / |Mx| * Mx, then project
  #pragma unroll
  for (int i = 0; i < 8; ++i) {
    float mx  = c[i];
    float tv  = mx + EPSF;
    float mxn = sqrtf(grp4_sum(tv * tv));
    float xv  = xpr[i] + EPSF;
    float xn  = sqrtf(grp4_sum(xv * xv));
    float arg = mxn / xn * atanh_ref(xn);
    float r   = tanhf(arg) / mxn * mx;
    float ssq = grp4_sum(r * r);
    res[i] = r * (1.0f - PROJ_EPSF) / fmaxf(ssq, 1.0f - PROJ_EPSF);
  }
}

// Phase 1: Ux for all (b,s) -> stored in d_out (same shape as output)
__global__ __launch_bounds__(256) void ux_kernel(
    const float* __restrict__ x, const float* __restrict__ U2,
    float* __restrict__ out)
{
  extern __shared__ float smem[];
  int tid = threadIdx.x, lane = tid & 31, wave = tid >> 5;
  size_t row0 = (size_t)blockIdx.x * 16;          // flattened (b*S + s)
  int n0 = blockIdx.y * 128 + wave * 16;
  float r8[8];
  mmm_tile(x + row0 * HD, U2, smem, tid, lane, n0, r8);
  int mrow = (lane >> 4) * 8, nn = n0 + (lane & 15);
  #pragma unroll
  for (int i = 0; i < 8; ++i)
    out[(row0 + mrow + i) * HD + nn] = r8[i];
}

// Phase 2 (per timestep): h_t = mob_add(mob_add(mmm(W2,h), Ux_t), b)
__global__ __launch_bounds__(256) void step_kernel(
    const float* __restrict__ h_in, const float* __restrict__ W2,
    const float* __restrict__ bvec, float* __restrict__ out,
    float* __restrict__ h_out, int t)
{
  extern __shared__ float smem[];
  int tid = threadIdx.x, lane = tid & 31, wave = tid >> 5;
  int row0 = blockIdx.x * 16;                      // batch tile
  int n0 = blockIdx.y * 128 + wave * 16;
  float wh[8];
  mmm_tile(h_in + (size_t)row0 * HD, W2, smem, tid, lane, n0, wh);

  int mrow = (lane >> 4) * 8, nn = n0 + (lane & 15);
  float bv  = bvec[nn] + EPSF;
  float nvb = grp4_sum(bv * bv);

  #pragma unroll
  for (int i = 0; i < 8; ++i) {
    int b = row0 + mrow + i;
    size_t oidx = ((size_t)b * SQ + t) * HD + nn;
    // mob_add(Wh, Ux)
    float u = wh[i];
    float v = out[oidx] + EPSF;                    // Ux from phase 1 (in-place)
    float nuv = 2.0f * grp4_sum(u * v);
    float nu  = grp4_sum(u * u);
    float nv  = grp4_sum(v * v);
    float den = 1.0f + nuv + nu * nv;
    float r = (1.0f + nuv + nv) / den * u + (1.0f - nu) / den * v;
    float ssq = grp4_sum(r * r);
    r *= (1.0f - PROJ_EPSF) / fmaxf(ssq, 1.0f - PROJ_EPSF);
    // mob_add(., b)
    float nuv2 = 2.0f * grp4_sum(r * bv);
    float nu2  = grp4_sum(r * r);
    float den2 = 1.0f + nuv2 + nu2 * nvb;
    float r2 = (1.0f + nuv2 + nvb) / den2 * r + (1.0f - nu2) / den2 * bv;
    float ssq2 = grp4_sum(r2 * r2);
    r2 *= (1.0f - PROJ_EPSF) / fmaxf(ssq2, 1.0f - PROJ_EPSF);

    out[oidx] = r2;                                // overwrite Ux_t with h_t
    h_out[(size_t)b * HD + nn] = r2;               // carry state
  }
}

extern "C" void kernel_launch(void* const* d_in, const int* in_sizes, int n_in,
                              void* d_out, int out_size, void* d_ws, size_t ws_size,
                              hipStream_t stream) {
  const float* x = (const float*)d_in[0];   // [B,S,H,D]
  const float* w = (const float*)d_in[1];   // [H*D, H*D]
  const float* u = (const float*)d_in[2];   // [H*D, H*D]
  const float* b = (const float*)d_in[3];   // [H*D]
  float* out = (float*)d_out;

  float* h0 = (float*)d_ws;                 // state double-buffer: 2 x 1 MB
  float* h1 = h0 + (size_t)NB * HD;
  hipMemsetAsync(h0, 0, (size_t)NB * HD * sizeof(float), stream);

  const size_t shmem = (size_t)KCH * 16 * sizeof(float);   // 32 KB

  // Phase 1: all Ux in parallel (65536 x 1024 x 1024 GEMM + epilogue)
  ux_kernel<<<dim3((NB * SQ) / 16, HD / 128), dim3(256), shmem, stream>>>(x, u, out);

  // Phase 2: sequential recurrence, one fused kernel per step
  for (int t = 0; t < SQ; ++t) {
    const float* hin = (t & 1) ? h1 : h0;
    float*      hout = (t & 1) ? h0 : h1;
    step_kernel<<<dim3(NB / 16, HD / 128), dim3(256), shmem, stream>>>(
        hin, w, b, out, hout, t);
  }
}